// SSIMLossForSequence_26774826123867
// MI455X (gfx1250) — compile-verified
//
#include <hip/hip_runtime.h>

typedef __attribute__((ext_vector_type(2))) float v2f;
typedef __attribute__((ext_vector_type(8))) float v8f;

#define L_SEQ 65536
#define B_SEQ 256
#define CHUNK 384              // 8 waves * 48 positions per block
#define EXT   400              // CHUNK + 5 (left halo) + 11 (right halo incl. K padding)
#define NPOS_PER_WAVE 48
#define GRID_X ((L_SEQ + CHUNK - 1) / CHUNK)   // 171
#define WTBL  64               // zero-padded band-weight table, index = d + 16, d in [-15,31]

__global__ __launch_bounds__(256)
void ssim_conv_kernel(const float* __restrict__ seq1,
                      const float* __restrict__ seq2,
                      float* __restrict__ partials) {
    __shared__ __align__(16) float sm_p[5 * EXT];   // 5 product channels + halo
    __shared__ float sm_conv[CHUNK * 5];            // conv outputs [pos][channel]
    __shared__ float sm_w[WTBL];                    // banded Gaussian weights, zero-padded
    __shared__ float sm_red[256];

    const int tid  = threadIdx.x;
    const int lane = tid & 31;
    const int wv   = tid >> 5;
    const int row  = blockIdx.y;
    const int chunkbase = blockIdx.x * CHUNK;
    const long rowoff = (long)row * L_SEQ;

    // ---- One-time weight table: sm_w[d+16] = w[d] for d in [0,10], else 0 ----
    // w[d] = exp(-(d-5)^2/2) / sqrt(2*pi)  (normalized 11-tap Gaussian window)
    if (tid < WTBL) {
        int d = tid - 16;
        float t = (float)(d - 5);
        float v = expf(-0.5f * t * t) * 0.3989422804014327f;
        sm_w[tid] = ((unsigned)d <= 10u) ? v : 0.0f;   // branchless cndmask
    }

    // ---- Stage product channels (with zero-padded halo) into LDS ----
    for (int i = tid; i < EXT; i += 256) {
        int x = chunkbase - 5 + i;
        bool ok = (x >= 0) && (x < L_SEQ);
        float a = ok ? seq1[rowoff + x] : 0.0f;
        float b = ok ? seq2[rowoff + x] : 0.0f;
        sm_p[0 * EXT + i] = a;
        sm_p[1 * EXT + i] = b;
        sm_p[2 * EXT + i] = a * a;
        sm_p[3 * EXT + i] = b * b;
        sm_p[4 * EXT + i] = a * b;
    }
    __syncthreads();

    // ---- Banded-GEMM convolution via 8 chained V_WMMA_F32_16X16X4_F32 ----
    // Row m = t*5 + c  (t = position tile 0..2, c = channel 0..4), row 15 unused.
    const int mRow = lane & 15;
    const int kOff = (lane >> 4) << 1;       // lanes 0-15: K+{0,1}; lanes 16-31: K+{2,3}
    const int mm   = (mRow == 15) ? 0 : mRow;
    const int tA   = mm / 5;
    const int cA   = mm - tA * 5;
    const int abase = cA * EXT + wv * NPOS_PER_WAVE + tA * 16;  // even -> 8B aligned
    const int n    = lane & 15;              // output position within tile (B/D column)
    const int wbase = kOff - n + 16;         // table index for kc=0, element 0

    v8f acc = {};
#pragma unroll
    for (int kc = 0; kc < 8; ++kc) {
        const int p0 = kc * 4 + kOff;
        // A operand: two consecutive product values for this row's segment (ds_load_b64)
        v2f a = *(const v2f*)&sm_p[abase + p0];
        // B operand: constant banded Gaussian matrix W[p][n] = w[p-n], from LDS table
        v2f b;
        b.x = sm_w[wbase + kc * 4];          // = band(p0 - n)
        b.y = sm_w[wbase + kc * 4 + 1];      // = band(p0 + 1 - n)
        acc = __builtin_amdgcn_wmma_f32_16x16x4_f32(
            /*neg_a=*/false, a, /*neg_b=*/false, b,
            /*c_mod=*/(short)0, acc, /*reuse_a=*/false, /*reuse_b=*/false);
    }

    // ---- Scatter D (15 used rows) into the LDS conv buffer ----
    const int mAdd = (lane >> 4) << 3;       // D: VGPR r holds rows r and r+8
#pragma unroll
    for (int r = 0; r < 8; ++r) {
        int m = r + mAdd;
        if (m != 15) {
            int td = m / 5;
            int cd = m - td * 5;
            sm_conv[(wv * NPOS_PER_WAVE + td * 16 + n) * 5 + cd] = acc[r];
        }
    }
    __syncthreads();

    // ---- Pointwise SSIM map + block-local sum ----
    const float C1 = 1.0e-4f;   // 0.01^2
    const float C2 = 9.0e-4f;   // 0.03^2
    float s = 0.0f;
    for (int p = tid; p < CHUNK; p += 256) {
        if (chunkbase + p < L_SEQ) {
            float mu1 = sm_conv[p * 5 + 0];
            float mu2 = sm_conv[p * 5 + 1];
            float e11 = sm_conv[p * 5 + 2];
            float e22 = sm_conv[p * 5 + 3];
            float e12 = sm_conv[p * 5 + 4];
            float mu1s = mu1 * mu1;
            float mu2s = mu2 * mu2;
            float m12  = mu1 * mu2;
            float sg1  = e11 - mu1s;
            float sg2  = e22 - mu2s;
            float sg12 = e12 - m12;
            float num = (2.0f * m12 + C1) * (2.0f * sg12 + C2);
            float den = (mu1s + mu2s + C1) * (sg1 + sg2 + C2);
            s += num / den;
        }
    }
    sm_red[tid] = s;
    __syncthreads();
    for (int off = 128; off > 0; off >>= 1) {
        if (tid < off) sm_red[tid] += sm_red[tid + off];
        __syncthreads();
    }
    if (tid == 0) partials[blockIdx.y * gridDim.x + blockIdx.x] = sm_red[0];
}

__global__ __launch_bounds__(256)
void ssim_reduce_kernel(const float* __restrict__ partials, int n,
                        float* __restrict__ out) {
    __shared__ float sm[256];
    float s = 0.0f;
    for (int i = threadIdx.x; i < n; i += 256) s += partials[i];
    sm[threadIdx.x] = s;
    __syncthreads();
    for (int off = 128; off > 0; off >>= 1) {
        if (threadIdx.x < off) sm[threadIdx.x] += sm[threadIdx.x + off];
        __syncthreads();
    }
    // mean over B*L = 256*65536 = 2^24 elements; loss = 1 - mean
    if (threadIdx.x == 0) out[0] = 1.0f - sm[0] * (1.0f / 16777216.0f);
}

extern "C" void kernel_launch(void* const* d_in, const int* in_sizes, int n_in,
                              void* d_out, int out_size, void* d_ws, size_t ws_size,
                              hipStream_t stream) {
    const float* s1 = (const float*)d_in[0];
    const float* s2 = (const float*)d_in[1];
    float* out    = (float*)d_out;
    float* parts  = (float*)d_ws;   // GRID_X * B_SEQ floats = 171 KB of scratch

    dim3 grid(GRID_X, B_SEQ);
    ssim_conv_kernel<<<grid, 256, 0, stream>>>(s1, s2, parts);
    ssim_reduce_kernel<<<1, 256, 0, stream>>>(parts, GRID_X * B_SEQ, out);
}